// LaneLoss_27135603376494
// MI455X (gfx1250) — compile-verified
//
#include <hip/hip_runtime.h>
#include <math.h>

#define N_IMG 4096
#define NA 32         // anchors
#define NK 72         // coords per row (W-1)
#define NH 16         // label rows
#define WROW 73       // 1 flag + 72 coords
#define IMG_SCALE 799.0f
#define WPB 4         // waves (images) per block
#define NBLOCKS (N_IMG / WPB)   // 1024

typedef float v2f __attribute__((ext_vector_type(2)));
typedef float v8f __attribute__((ext_vector_type(8)));
typedef unsigned int v4u __attribute__((ext_vector_type(4)));
typedef int v4i __attribute__((ext_vector_type(4)));
typedef int v8i __attribute__((ext_vector_type(8)));

#if defined(__gfx1250__) && __has_builtin(__builtin_amdgcn_tensor_load_to_lds)
#define USE_TDM 1
#else
#define USE_TDM 0
#endif

#if USE_TDM
// One TDM DMA of `nelem` contiguous f32 from global -> LDS (1-row 2D tile).
// D# layout per CDNA5 ISA ch.8: group0 = {flags, lds_addr, global_addr, type=2},
// group1 = {data_size=4B, tensor_dim0/1, tile_dim0/1/2, strides}.
__device__ __forceinline__ void tdm_load_1d(unsigned lds_off, const void* gptr,
                                            unsigned nelem) {
  unsigned long long ga = (unsigned long long)gptr;
  v4u g0;
  g0[0] = 1u;                                                  // count=1, user D#
  g0[1] = lds_off;                                             // lds_addr (bytes)
  g0[2] = (unsigned)(ga & 0xFFFFFFFFu);                        // global_addr[31:0]
  g0[3] = ((unsigned)((ga >> 32) & 0x01FFFFFFu)) | (2u << 30); // addr[56:32] | type=2
  v8i g1;
  g1[0] = (int)(2u << 16);                                     // data_size=2 (4B), mask=0
  g1[1] = (int)((nelem & 0xFFFFu) << 16);                      // tensor_dim0[15:0] @bits63:48
  g1[2] = (int)(((nelem >> 16) & 0xFFFFu) | (1u << 16));       // tensor_dim0[31:16] | tensor_dim1=1
  g1[3] = (int)((nelem & 0xFFFFu) << 16);                      // tensor_dim1 hi=0 | tile_dim0
  g1[4] = 1;                                                   // tile_dim1=1, tile_dim2=0
  g1[5] = (int)nelem;                                          // tensor_dim0_stride lo
  g1[6] = (int)((nelem & 0xFFFFu) << 16);                      // stride hi=0 | dim1_stride lo16
  g1[7] = 0;                                                   // dim1_stride hi
  v4i gz = {0, 0, 0, 0};
#if __has_include(<hip/amd_detail/amd_gfx1250_TDM.h>)          // therock/clang-23: 6-arg form
  v8i gz8 = {0, 0, 0, 0, 0, 0, 0, 0};
  __builtin_amdgcn_tensor_load_to_lds(g0, g1, gz, gz, gz8, 0);
#else                                                          // ROCm 7.2/clang-22: 5-arg form
  __builtin_amdgcn_tensor_load_to_lds(g0, g1, gz, gz, 0);
#endif
}
#endif

__global__ __launch_bounds__(WPB * 32)
void lane_loss_main(const float* __restrict__ cls_exit,
                    const float* __restrict__ point,
                    const float* __restrict__ label,
                    float* __restrict__ partials) {
  __shared__ __attribute__((aligned(16))) float sPre[WPB][NA * NK];   // raw point, 36864 B
  __shared__ __attribute__((aligned(16))) float sLab[WPB][NH * WROW]; // raw label, 18688 B
  __shared__ __attribute__((aligned(16))) float sD[WPB][NH * NA];     // dists,      8192 B
  __shared__ float sAcc[WPB][3];

  const int lane = threadIdx.x & 31;
  const int wid  = threadIdx.x >> 5;
  const int n    = blockIdx.x * WPB + wid;

#if USE_TDM
  // ---- TDM: DMA this wave's image tiles straight into LDS (no VGPR staging) ----
  tdm_load_1d((unsigned)(unsigned long long)(uintptr_t)&sPre[wid][0],
              point + (size_t)n * NA * NK, NA * NK);
  tdm_load_1d((unsigned)(unsigned long long)(uintptr_t)&sLab[wid][0],
              label + (size_t)n * NH * WROW, NH * WROW);
  __builtin_amdgcn_s_wait_tensorcnt(0);
#else
  // ---- fallback staging: b128 loads -> LDS (raw, unscaled) ----
  {
    const float4* gp = (const float4*)(point + (size_t)n * NA * NK);
    float4* lp = (float4*)sPre[wid];
#pragma unroll
    for (int i = 0; i < (NA * NK / 4) / 32; ++i) lp[lane + i * 32] = gp[lane + i * 32];
    const float4* gl = (const float4*)(label + (size_t)n * NH * WROW);
    float4* ll = (float4*)sLab[wid];
    for (int i = lane; i < (NH * WROW) / 4; i += 32) ll[i] = gl[i];
  }
#endif
  __syncthreads();

  // ---- dists[h][a] = sum_k |799*p[a][k] - lab[h][k]| via WMMA f32 16x16x4 ----
  // A[m][kk] = absdiff for a 16-anchor tile, B = ones(4x16) -> D holds row sums.
  // A layout: lanes 0-15 hold K=0,1 for M=lane; lanes 16-31 hold K=2,3 for M=lane-16.
  // Two label rows per iteration -> two independent accumulator chains (ILP)
  // while #pragma unroll 1 keeps VGPRs < 256 (no s_set_vgpr_msb churn).
  const int half = lane >> 4;     // 0 or 1
  const int mrow = lane & 15;
  v2f ones; ones.x = 1.0f; ones.y = 1.0f;

#pragma unroll 1
  for (int tile = 0; tile < 2; ++tile) {
    // this lane's 36 pre elements for the tile, scaled once into registers
    float pr[2 * (NK / 4)];
    {
      const float* prow = &sPre[wid][(tile * 16 + mrow) * NK];
#pragma unroll
      for (int c = 0; c < NK / 4; ++c) {
        const int k = 4 * c + 2 * half;
        pr[2 * c]     = prow[k]     * IMG_SCALE;
        pr[2 * c + 1] = prow[k + 1] * IMG_SCALE;
      }
    }
#pragma unroll 1
    for (int h = 0; h < NH; h += 2) {
      const float* lrow0 = &sLab[wid][h * WROW + 1];
      const float* lrow1 = &sLab[wid][(h + 1) * WROW + 1];
      v8f acc0 = {};
      v8f acc1 = {};
#pragma unroll
      for (int c = 0; c < NK / 4; ++c) {          // 2x18 chained WMMAs, f32 accum
        const int k = 4 * c + 2 * half;
        v2f a0, a1;
        a0.x = fabsf(pr[2 * c]     - lrow0[k]);
        a0.y = fabsf(pr[2 * c + 1] - lrow0[k + 1]);
        a1.x = fabsf(pr[2 * c]     - lrow1[k]);
        a1.y = fabsf(pr[2 * c + 1] - lrow1[k + 1]);
        acc0 = __builtin_amdgcn_wmma_f32_16x16x4_f32(
            false, a0, false, ones, (short)0, acc0, false, false);
        acc1 = __builtin_amdgcn_wmma_f32_16x16x4_f32(
            false, a1, false, ones, (short)0, acc1, false, false);
      }
      // D VGPR j: lanes0-15 = row j, lanes16-31 = row j+8 (replicated across cols)
      if (mrow == 0) {
        float* dst0 = &sD[wid][h * NA + tile * 16 + 8 * half];
        float* dst1 = &sD[wid][(h + 1) * NA + tile * 16 + 8 * half];
        float4 t;
        t.x = acc0[0]; t.y = acc0[1]; t.z = acc0[2]; t.w = acc0[3]; *(float4*)dst0 = t;
        t.x = acc0[4]; t.y = acc0[5]; t.z = acc0[6]; t.w = acc0[7]; *(float4*)(dst0 + 4) = t;
        t.x = acc1[0]; t.y = acc1[1]; t.z = acc1[2]; t.w = acc1[3]; *(float4*)dst1 = t;
        t.x = acc1[4]; t.y = acc1[5]; t.z = acc1[6]; t.w = acc1[7]; *(float4*)(dst1 + 4) = t;
      }
    }
  }
  __syncthreads();

  // ---- greedy matching: lane = anchor; 16 sequential wave (min,argmin-first) ----
  float dh[NH];
#pragma unroll
  for (int h = 0; h < NH; ++h) dh[h] = sD[wid][h * NA + lane];

  bool matched = false;
  float point_sum = 0.0f;   // uniform across lanes
  float lane_cnt  = 0.0f;   // uniform across lanes
#pragma unroll
  for (int h = 0; h < NH; ++h) {
    float mv = matched ? 3.0e38f : dh[h];
    int   mi = lane;
#pragma unroll
    for (int off = 16; off >= 1; off >>= 1) {
      float ov = __shfl_xor(mv, off, 32);
      int   oi = __shfl_xor(mi, off, 32);
      if (ov < mv || (ov == mv && oi < mi)) { mv = ov; mi = oi; }
    }
    const bool valid = sLab[wid][h * WROW] != 0.0f;
    if (valid) {
      if (lane == mi) matched = true;   // all lanes agree on mi
      point_sum += mv;                  // mean-L1 * 72 of the matched anchor
      lane_cnt  += 1.0f;
    }
  }

  // ---- focal NLL: target = matched bit for this lane's anchor ----
  const int tg = matched ? 1 : 0;
  const float p  = cls_exit[(size_t)n * NA * 2 + lane * 2 + tg];
  const float om = 1.0f - p;
  float focal = om * om * logf(p);
#pragma unroll
  for (int off = 16; off >= 1; off >>= 1) focal += __shfl_xor(focal, off, 32);

  if (lane == 0) {
    sAcc[wid][0] = focal;
    sAcc[wid][1] = point_sum;
    sAcc[wid][2] = lane_cnt;
  }
  __syncthreads();
  if (threadIdx.x == 0) {
    float f = 0.0f, ps = 0.0f, lc = 0.0f;
    for (int w = 0; w < WPB; ++w) { f += sAcc[w][0]; ps += sAcc[w][1]; lc += sAcc[w][2]; }
    partials[blockIdx.x * 3 + 0] = f;
    partials[blockIdx.x * 3 + 1] = ps;
    partials[blockIdx.x * 3 + 2] = lc;
  }
}

__global__ __launch_bounds__(256)
void lane_loss_finalize(const float* __restrict__ partials, float* __restrict__ out) {
  __shared__ float red[256][3];
  const int tid = threadIdx.x;
  float f = 0.0f, ps = 0.0f, lc = 0.0f;
  for (int i = tid; i < NBLOCKS; i += 256) {   // fixed-order per-thread partials
    f  += partials[i * 3 + 0];
    ps += partials[i * 3 + 1];
    lc += partials[i * 3 + 2];
  }
  red[tid][0] = f; red[tid][1] = ps; red[tid][2] = lc;
  __syncthreads();
  for (int s = 128; s > 0; s >>= 1) {          // deterministic tree
    if (tid < s) {
      red[tid][0] += red[tid + s][0];
      red[tid][1] += red[tid + s][1];
      red[tid][2] += red[tid + s][2];
    }
    __syncthreads();
  }
  if (tid == 0) {
    const float cls_loss   = -red[0][0] / (float)(N_IMG * NA);
    const float point_loss =  red[0][1] / (float)NK;   // /72 (mean over coords)
    out[0] = cls_loss + 0.4f * point_loss / (red[0][2] + 1e-6f);
  }
}

extern "C" void kernel_launch(void* const* d_in, const int* in_sizes, int n_in,
                              void* d_out, int out_size, void* d_ws, size_t ws_size,
                              hipStream_t stream) {
  const float* cls_exit = (const float*)d_in[0];  // (4096, 32, 2)
  const float* point    = (const float*)d_in[1];  // (4096, 32, 72)
  const float* label    = (const float*)d_in[2];  // (4096, 16, 73)
  float* partials = (float*)d_ws;                 // NBLOCKS*3 floats = 12 KB
  lane_loss_main<<<NBLOCKS, WPB * 32, 0, stream>>>(cls_exit, point, label, partials);
  lane_loss_finalize<<<1, 256, 0, stream>>>(partials, (float*)d_out);
}